// Encoder_60318520705555
// MI455X (gfx1250) — compile-verified
//
#include <hip/hip_runtime.h>

#define BATCH 8
#define NPTS  2048
#define BN    16384      // BATCH * NPTS  (power of two: 2^14)
#define KNN   16

typedef __attribute__((ext_vector_type(2))) float v2f;
typedef __attribute__((ext_vector_type(8))) float v8f;

// ---------------------------------------------------------------------------
// xyz (B,N,3) row-major -> channel-major (4, BN), row 3 zero-padded so the
// WMMA KNN kernel can use K-steps of 4.
// ---------------------------------------------------------------------------
__global__ void xyzt_kernel(const float* __restrict__ x, float* __restrict__ xT) {
  int p = blockIdx.x * blockDim.x + threadIdx.x;
  if (p >= BN) return;
  xT[0 * BN + p] = x[p * 3 + 0];
  xT[1 * BN + p] = x[p * 3 + 1];
  xT[2 * BN + p] = x[p * 3 + 2];
  xT[3 * BN + p] = 0.0f;
}

// Per-point squared norm over C channels (channel-major features).
__global__ void sqnorm_kernel(const float* __restrict__ F, float* __restrict__ sq, int C) {
  int p = blockIdx.x * blockDim.x + threadIdx.x;
  if (p >= BN) return;
  float s = 0.f;
  for (int c = 0; c < C; ++c) { float v = F[(size_t)c * BN + p]; s += v * v; }
  sq[p] = s;
}

// ---------------------------------------------------------------------------
// Fused KNN: one wave per (batch, 16-query-row tile).
// Gram tile (16x16) accumulated with V_WMMA_F32_16X16X4_F32 over K=CPAD,
// A-fragments (query rows) cached in registers across all 128 column tiles.
// dist = |a|^2 + |b|^2 - 2 a.b ; lanes 0..15 keep a sorted top-16 in regs.
// f32 16x16x4 VGPR layout (ISA 7.12.2):
//   A 16x4: v0 = A[m=l][k0+2*half], v1 = A[m=l][k0+2*half+1]
//   B 4x16: v0 = B[k0+2*half][n=l], v1 = B[k0+2*half+1][n=l]
//   C/D   : vr = C[m=r+8*half][n=l]
// ---------------------------------------------------------------------------
template <int CPAD>
__global__ __launch_bounds__(32) void knn_kernel(const float* __restrict__ F,
                                                 const float* __restrict__ sq,
                                                 int* __restrict__ idxOut) {
  __shared__ float tile[16][17];
  const int lane = threadIdx.x;
  const int half = lane >> 4, l = lane & 15;
  const int b  = blockIdx.x >> 7;            // NPTS/16 = 128 row tiles per batch
  const int i0 = (blockIdx.x & 127) << 4;
  const size_t cb = (size_t)b * NPTS;
  constexpr int KS = CPAD / 4;

  v2f areg[KS];
#pragma unroll
  for (int s = 0; s < KS; ++s) {
    int k = 4 * s + 2 * half;
    areg[s].x = F[(size_t)k * BN + cb + i0 + l];
    areg[s].y = F[(size_t)(k + 1) * BN + cb + i0 + l];
  }
  float aa = sq[cb + i0 + l];

  float bestd[KNN];
  int   besti[KNN];
#pragma unroll
  for (int t = 0; t < KNN; ++t) { bestd[t] = 3.0e38f; besti[t] = 0; }

  for (int jt = 0; jt < NPTS / 16; ++jt) {
    const int j0 = jt << 4;
    v8f acc = {};
#pragma unroll
    for (int s = 0; s < KS; ++s) {
      int k = 4 * s + 2 * half;
      v2f bf;
      bf.x = F[(size_t)k * BN + cb + j0 + l];
      bf.y = F[(size_t)(k + 1) * BN + cb + j0 + l];
      acc = __builtin_amdgcn_wmma_f32_16x16x4_f32(false, areg[s], false, bf,
                                                  (short)0, acc, false, false);
    }
    __syncthreads();  // WAR vs previous iteration's tile reads
#pragma unroll
    for (int r = 0; r < 8; ++r) tile[r + 8 * half][l] = acc[r];
    __syncthreads();
    if (lane < 16) {
#pragma unroll
      for (int t = 0; t < 16; ++t) {
        int   j = j0 + t;
        float d = aa + sq[cb + j] - 2.0f * tile[lane][t];
        if (d < bestd[KNN - 1]) {
          bestd[KNN - 1] = d;
          besti[KNN - 1] = j;
#pragma unroll
          for (int u = KNN - 1; u > 0; --u) {  // register-only compare-swap
            bool  sw = bestd[u] < bestd[u - 1];
            float d0 = bestd[u - 1], d1 = bestd[u];
            int   e0 = besti[u - 1], e1 = besti[u];
            bestd[u - 1] = sw ? d1 : d0; bestd[u] = sw ? d0 : d1;
            besti[u - 1] = sw ? e1 : e0; besti[u] = sw ? e0 : e1;
          }
        }
      }
    }
  }
  if (lane < 16) {
    size_t base = ((size_t)b * NPTS + i0 + lane) * KNN;
#pragma unroll
    for (int t = 0; t < KNN; ++t) idxOut[base + t] = besti[t];
  }
}

// ---------------------------------------------------------------------------
// Gather 16 neighbors of raw xyz, subtract mean, 3x3 covariance, emit
// h0 = concat(xyz^T, cov9) as (12, BN).
// ---------------------------------------------------------------------------
__global__ void cov_kernel(const float* __restrict__ x, const int* __restrict__ idx,
                           float* __restrict__ h0) {
  int p = blockIdx.x * blockDim.x + threadIdx.x;
  if (p >= BN) return;
  int b = p >> 11, n = p & (NPTS - 1);
  const float* xb = x + (size_t)b * NPTS * 3;
  float nx[KNN], ny[KNN], nz[KNN];
  float mx = 0, my = 0, mz = 0;
#pragma unroll
  for (int k = 0; k < KNN; ++k) {
    int j = idx[(size_t)p * KNN + k];
    nx[k] = xb[j * 3 + 0]; ny[k] = xb[j * 3 + 1]; nz[k] = xb[j * 3 + 2];
    mx += nx[k]; my += ny[k]; mz += nz[k];
  }
  mx *= (1.f / KNN); my *= (1.f / KNN); mz *= (1.f / KNN);
  float c00 = 0, c01 = 0, c02 = 0, c11 = 0, c12 = 0, c22 = 0;
#pragma unroll
  for (int k = 0; k < KNN; ++k) {
    float dx = nx[k] - mx, dy = ny[k] - my, dz = nz[k] - mz;
    c00 += dx * dx; c01 += dx * dy; c02 += dx * dz;
    c11 += dy * dy; c12 += dy * dz; c22 += dz * dz;
  }
  h0[0 * BN + p] = xb[n * 3 + 0];
  h0[1 * BN + p] = xb[n * 3 + 1];
  h0[2 * BN + p] = xb[n * 3 + 2];
  h0[3 * BN + p] = c00; h0[4 * BN + p]  = c01; h0[5 * BN + p]  = c02;
  h0[6 * BN + p] = c01; h0[7 * BN + p]  = c11; h0[8 * BN + p]  = c12;
  h0[9 * BN + p] = c02; h0[10 * BN + p] = c12; h0[11 * BN + p] = c22;
}

// ---------------------------------------------------------------------------
// Y(M,BN) = W(M,Kd) * X(Kd,BN) + bias, via f32 WMMA.
// Register-blocked: each wave computes 64 rows (4 M-tiles) x 16 cols, loading
// ONE B fragment per k-step and reusing it across 4 WMMAs (4x less X traffic).
// 4 waves/block cover a 64x64 output block. global_prefetch (gfx1250) hides
// the B-row stream latency for the deep-K layers.
// ---------------------------------------------------------------------------
__global__ __launch_bounds__(128) void gemm_bias_kernel(const float* __restrict__ W,
                                                        const float* __restrict__ X,
                                                        const float* __restrict__ bias,
                                                        float* __restrict__ Y,
                                                        int M, int Kd) {
  const int wave = threadIdx.x >> 5;
  const int lane = threadIdx.x & 31;
  const int half = lane >> 4, l = lane & 15;
  const int n0 = ((blockIdx.x << 2) + wave) << 4;
  const int m0 = blockIdx.y << 6;  // 64 rows per grid.y step
  v8f acc0 = {}, acc1 = {}, acc2 = {}, acc3 = {};
  for (int k0 = 0; k0 < Kd; k0 += 4) {
    v2f bf;
    bf.x = X[(size_t)(k0 + 2 * half + 0) * BN + n0 + l];
    bf.y = X[(size_t)(k0 + 2 * half + 1) * BN + n0 + l];
    // prefetch B rows two k-steps ahead (global_prefetch_b8)
    int kp = (k0 + 8 < Kd) ? (k0 + 8) : k0;
    __builtin_prefetch(&X[(size_t)(kp + 2 * half) * BN + n0 + l], 0, 1);
    const size_t wrow = (size_t)(m0 + l) * Kd + k0 + 2 * half;
    v2f a0, a1, a2, a3;
    a0.x = W[wrow + (size_t)0  * Kd + 0]; a0.y = W[wrow + (size_t)0  * Kd + 1];
    a1.x = W[wrow + (size_t)16 * Kd + 0]; a1.y = W[wrow + (size_t)16 * Kd + 1];
    a2.x = W[wrow + (size_t)32 * Kd + 0]; a2.y = W[wrow + (size_t)32 * Kd + 1];
    a3.x = W[wrow + (size_t)48 * Kd + 0]; a3.y = W[wrow + (size_t)48 * Kd + 1];
    acc0 = __builtin_amdgcn_wmma_f32_16x16x4_f32(false, a0, false, bf, (short)0, acc0, false, false);
    acc1 = __builtin_amdgcn_wmma_f32_16x16x4_f32(false, a1, false, bf, (short)0, acc1, false, false);
    acc2 = __builtin_amdgcn_wmma_f32_16x16x4_f32(false, a2, false, bf, (short)0, acc2, false, false);
    acc3 = __builtin_amdgcn_wmma_f32_16x16x4_f32(false, a3, false, bf, (short)0, acc3, false, false);
  }
#pragma unroll
  for (int r = 0; r < 8; ++r) {
    int m = m0 + r + 8 * half;
    Y[(size_t)(m +  0) * BN + n0 + l] = acc0[r] + bias[m +  0];
    Y[(size_t)(m + 16) * BN + n0 + l] = acc1[r] + bias[m + 16];
    Y[(size_t)(m + 32) * BN + n0 + l] = acc2[r] + bias[m + 32];
    Y[(size_t)(m + 48) * BN + n0 + l] = acc3[r] + bias[m + 48];
  }
}

// Per-channel batch-stat reduction over all BN columns (axis (0,2) of ref).
__global__ __launch_bounds__(256) void bn_stats_kernel(const float* __restrict__ Y,
                                                       float* __restrict__ mu,
                                                       float* __restrict__ rs) {
  __shared__ float ss[256], s2[256];
  int c = blockIdx.x, t = threadIdx.x;
  const float* row = Y + (size_t)c * BN;
  float a = 0.f, b = 0.f;
  for (int i = t; i < BN; i += 256) { float v = row[i]; a += v; b += v * v; }
  ss[t] = a; s2[t] = b; __syncthreads();
  for (int o = 128; o > 0; o >>= 1) {
    if (t < o) { ss[t] += ss[t + o]; s2[t] += s2[t + o]; }
    __syncthreads();
  }
  if (t == 0) {
    float m = ss[0] * (1.f / BN);
    float v = s2[0] * (1.f / BN) - m * m;
    mu[c] = m;
    rs[c] = rsqrtf(v + 1e-5f);
  }
}

__global__ void bn_apply_kernel(float* __restrict__ Y, const float* __restrict__ mu,
                                const float* __restrict__ rs, const float* __restrict__ gm,
                                const float* __restrict__ bt, int C, int relu) {
  size_t total = (size_t)C * BN;
  for (size_t t = (size_t)blockIdx.x * blockDim.x + threadIdx.x; t < total;
       t += (size_t)gridDim.x * blockDim.x) {
    int   c = (int)(t >> 14);  // BN == 2^14
    float v = (Y[t] - mu[c]) * rs[c] * gm[c] + bt[c];
    Y[t] = relu ? fmaxf(v, 0.f) : v;
  }
}

// out[c,p] = max over k of F[c, b*N + idx[p,k]]   (channel-major in/out)
__global__ void gathermax_kernel(const float* __restrict__ F, const int* __restrict__ idx,
                                 float* __restrict__ out, int C) {
  size_t total = (size_t)C * BN;
  for (size_t t = (size_t)blockIdx.x * blockDim.x + threadIdx.x; t < total;
       t += (size_t)gridDim.x * blockDim.x) {
    int c = (int)(t >> 14);
    int p = (int)(t & (BN - 1));
    int b = p >> 11;
    const float* Fc = F + (size_t)c * BN + (size_t)b * NPTS;
    const int*   ip = idx + (size_t)p * KNN;
    float m = -3.0e38f;
#pragma unroll
    for (int k = 0; k < KNN; ++k) m = fmaxf(m, Fc[ip[k]]);
    out[t] = m;
  }
}

// Global max over N per (b,c); blockIdx.x = b*512 + c == flat output index.
__global__ __launch_bounds__(256) void maxpool_kernel(const float* __restrict__ Y,
                                                      float* __restrict__ out) {
  __shared__ float sm[256];
  int b = blockIdx.x >> 9, c = blockIdx.x & 511;
  const float* row = Y + (size_t)c * BN + (size_t)b * NPTS;
  float m = -3.0e38f;
  for (int i = threadIdx.x; i < NPTS; i += 256) m = fmaxf(m, row[i]);
  sm[threadIdx.x] = m; __syncthreads();
  for (int o = 128; o > 0; o >>= 1) {
    if (threadIdx.x < o) sm[threadIdx.x] = fmaxf(sm[threadIdx.x], sm[threadIdx.x + o]);
    __syncthreads();
  }
  if (threadIdx.x == 0) out[blockIdx.x] = sm[0];
}

// ---------------------------------------------------------------------------
extern "C" void kernel_launch(void* const* d_in, const int* in_sizes, int n_in,
                              void* d_out, int out_size, void* d_ws, size_t ws_size,
                              hipStream_t stream) {
  const float* x   = (const float*)d_in[0];
  const float *W1  = (const float*)d_in[1],  *b1  = (const float*)d_in[2],
              *gm1 = (const float*)d_in[3],  *bt1 = (const float*)d_in[4];
  const float *W2  = (const float*)d_in[5],  *b2  = (const float*)d_in[6],
              *gm2 = (const float*)d_in[7],  *bt2 = (const float*)d_in[8];
  const float *W3  = (const float*)d_in[9],  *b3  = (const float*)d_in[10],
              *gm3 = (const float*)d_in[11], *bt3 = (const float*)d_in[12];
  const float *Wg1 = (const float*)d_in[13], *bg1 = (const float*)d_in[14],
              *gmg1= (const float*)d_in[15], *btg1= (const float*)d_in[16];
  const float *Wg2 = (const float*)d_in[17], *bg2 = (const float*)d_in[18],
              *gmg2= (const float*)d_in[19], *btg2= (const float*)d_in[20];
  const float *W4  = (const float*)d_in[21], *b4  = (const float*)d_in[22],
              *gm4 = (const float*)d_in[23], *bt4 = (const float*)d_in[24];

  char*  ws  = (char*)d_ws;
  size_t off = 0;
  auto alloc = [&](size_t bytes) -> char* {
    char* p = ws + off;
    off = (off + bytes + 255) & ~(size_t)255;
    return p;
  };
  float* xT   = (float*)alloc((size_t)4 * BN * 4);
  float* sq   = (float*)alloc((size_t)BN * 4);
  int*   idx  = (int*)  alloc((size_t)BN * KNN * 4);
  float* mu   = (float*)alloc(1024 * 4);
  float* rs   = (float*)alloc(1024 * 4);
  float* f1   = (float*)alloc((size_t)64 * BN * 4);
  float* f2   = (float*)alloc((size_t)64 * BN * 4);
  float* mid1 = (float*)alloc((size_t)128 * BN * 4);
  float* mid2 = (float*)alloc((size_t)128 * BN * 4);
  float* big1 = (float*)alloc((size_t)1024 * BN * 4);
  float* big2 = (float*)alloc((size_t)512 * BN * 4);
  float* h0   = mid1;  // 12 x BN, mid1 is free at this stage

  auto conv = [&](const float* W, const float* bb, const float* gm, const float* bt,
                  const float* X, float* Y, int M, int Kd, int relu) {
    gemm_bias_kernel<<<dim3(BN / 64, M / 64), 128, 0, stream>>>(W, X, bb, Y, M, Kd);
    bn_stats_kernel<<<M, 256, 0, stream>>>(Y, mu, rs);
    bn_apply_kernel<<<dim3((M * BN) / 256), 256, 0, stream>>>(Y, mu, rs, gm, bt, M, relu);
  };

  // Stage 0: xyz KNN + covariance features -> h0 (12, BN)
  xyzt_kernel<<<BN / 256, 256, 0, stream>>>(x, xT);
  sqnorm_kernel<<<BN / 256, 256, 0, stream>>>(xT, sq, 4);
  knn_kernel<4><<<BATCH * (NPTS / 16), 32, 0, stream>>>(xT, sq, idx);
  cov_kernel<<<BN / 256, 256, 0, stream>>>(x, idx, h0);

  // conv1..3
  conv(W1, b1, gm1, bt1, h0, f1, 64, 12, 1);
  conv(W2, b2, gm2, bt2, f1, f2, 64, 64, 1);
  conv(W3, b3, gm3, bt3, f2, f1, 64, 64, 1);

  // graph layer 1 (C=64 -> 128)
  sqnorm_kernel<<<BN / 256, 256, 0, stream>>>(f1, sq, 64);
  knn_kernel<64><<<BATCH * (NPTS / 16), 32, 0, stream>>>(f1, sq, idx);
  gathermax_kernel<<<dim3((64 * BN) / 256), 256, 0, stream>>>(f1, idx, f2, 64);
  conv(Wg1, bg1, gmg1, btg1, f2, mid1, 128, 64, 1);

  // graph layer 2 (C=128 -> 1024)
  sqnorm_kernel<<<BN / 256, 256, 0, stream>>>(mid1, sq, 128);
  knn_kernel<128><<<BATCH * (NPTS / 16), 32, 0, stream>>>(mid1, sq, idx);
  gathermax_kernel<<<dim3((128 * BN) / 256), 256, 0, stream>>>(mid1, idx, mid2, 128);
  conv(Wg2, bg2, gmg2, btg2, mid2, big1, 1024, 128, 1);

  // conv4 (1024 -> 512, no relu) + global max-pool
  conv(W4, b4, gm4, bt4, big1, big2, 512, 1024, 0);
  maxpool_kernel<<<BATCH * 512, 256, 0, stream>>>(big2, (float*)d_out);

  (void)in_sizes; (void)n_in; (void)out_size; (void)ws_size;
}